// JiTVMamba_22316650070180
// MI455X (gfx1250) — compile-verified
//
#include <hip/hip_runtime.h>

#define BB 4
#define LL 256
#define HD 384
#define MH_ 1536
#define RANK_ 24
#define NST 16

typedef __attribute__((ext_vector_type(2))) float v2f;
typedef __attribute__((ext_vector_type(4))) float v4f;
typedef __attribute__((ext_vector_type(8))) float v8f;

// ---------------------------------------------------------------------------
// Batched/strided WMMA fp32 GEMM:  C[m,n] = sum_k A[m,k]*W[n,k] + bias[n]
// Each wave computes a 16(M) x 64(N) strip: one A fragment feeds 4 WMMAs.
// Fragment layouts per CDNA5 ISA 7.12.2 (32-bit, wave32):
//   A 16x4 : lane r=lane&15 -> M=r ; k = (lane>>4)*2 + vgpr
//   B 4x16 : lane r=lane&15 -> N=r ; k = (lane>>4)*2 + vgpr
//   C/D    : M = (lane>>4)*8 + vgpr ; N = lane&15
// W is always row-major over k (lwk==1). AK1: A contiguous over k (lak==1).
// NAL: N is a multiple of 64 -> no bounds masking in the hot loop.
// Batch offset: ptr += (z % bmod)*s_lo + (z / bmod)*s_hi
// ---------------------------------------------------------------------------
template <int AK1, int NAL>
__global__ void __launch_bounds__(128)
gemm_wmma_f32(const float* __restrict__ A, const float* __restrict__ W,
              const float* __restrict__ bias, float* __restrict__ C,
              int M, int N, int K,
              long lam, long lak, long lwn, long lcm, long lcn,
              int bA, long sAlo, long sAhi,
              int bW, long sWlo, long sWhi,
              int bC, long sClo, long sChi)
{
    int z = blockIdx.z;
    A += (long)(z % bA) * sAlo + (long)(z / bA) * sAhi;
    W += (long)(z % bW) * sWlo + (long)(z / bW) * sWhi;
    C += (long)(z % bC) * sClo + (long)(z / bC) * sChi;

    int lane = threadIdx.x & 31;
    int wave = threadIdx.x >> 5;            // 4 waves -> 4 M-tiles of the 64-row block
    int tm  = blockIdx.x * 64 + wave * 16;
    int tn0 = blockIdx.y * 64;

    int r  = lane & 15;
    int ks = (lane >> 4) << 1;              // 0 or 2
    long arow = (long)(tm + r) * lam;

    int  nc[4];
    bool nv[4];
    long wrow[4];
#pragma unroll
    for (int j = 0; j < 4; ++j) {
        nc[j] = tn0 + j * 16 + r;
        nv[j] = NAL ? true : (nc[j] < N);
        wrow[j] = (long)(NAL ? nc[j] : (nv[j] ? nc[j] : 0)) * lwn;
    }

    v8f acc[4];
#pragma unroll
    for (int j = 0; j < 4; ++j) acc[j] = (v8f)0.f;

    for (int kb = 0; kb < K; kb += 4) {
        v2f a;
        if (AK1) {
            a = *(const v2f*)(A + arow + kb + ks);      // one b64 load
        } else {
            a.x = A[arow + (long)(kb + ks) * lak];
            a.y = A[arow + (long)(kb + ks + 1) * lak];
        }
#pragma unroll
        for (int j = 0; j < 4; ++j) {
            v2f wv = *(const v2f*)(W + wrow[j] + kb + ks);  // one b64 load
            v2f b;
            if (NAL) {
                b = wv;
            } else {
                b.x = nv[j] ? wv.x : 0.f;
                b.y = nv[j] ? wv.y : 0.f;
            }
            acc[j] = __builtin_amdgcn_wmma_f32_16x16x4_f32(false, a, false, b,
                                                           (short)0, acc[j], false, false);
        }
    }

    int mb = tm + ((lane >> 4) << 3);
#pragma unroll
    for (int j = 0; j < 4; ++j) {
        if (nv[j]) {
            float bv = bias ? bias[nc[j]] : 0.f;
#pragma unroll
            for (int v = 0; v < 8; ++v)
                C[(long)(mb + v) * lcm + (long)nc[j] * lcn] = acc[j][v] + bv;
        }
    }
}

// ---------------------------------------------------------------------------
// Small helpers
// ---------------------------------------------------------------------------
__device__ __forceinline__ float siluf(float x) { return x / (1.f + __expf(-x)); }

__device__ __forceinline__ float blockReduceSum128(float v, float* sh)
{
    int t = threadIdx.x;
    sh[t] = v; __syncthreads();
#pragma unroll
    for (int s = 64; s > 0; s >>= 1) {
        if (t < s) sh[t] += sh[t + s];
        __syncthreads();
    }
    float r = sh[0]; __syncthreads();
    return r;
}

// ---------------------------------------------------------------------------
// Time/class embedding -> c (B,384)
// ---------------------------------------------------------------------------
__global__ void __launch_bounds__(128)
time_embed_k(const float* __restrict__ t, const int* __restrict__ y,
             const float* __restrict__ w1, const float* __restrict__ b1,
             const float* __restrict__ w2, const float* __restrict__ b2,
             const float* __restrict__ ytab, float* __restrict__ c)
{
    int b = blockIdx.x, tid = threadIdx.x;
    __shared__ float te[256];
    __shared__ float hid[HD];
    float tv = t[b];
    for (int i = tid; i < 256; i += 128) {
        int f = i & 127;
        float fr = __expf(-logf(10000.f) * (float)f / 128.f);
        float a = tv * fr;
        te[i] = (i < 128) ? cosf(a) : sinf(a);
    }
    __syncthreads();
    for (int j = tid; j < HD; j += 128) {
        float s = b1[j];
        for (int k = 0; k < 256; ++k) s += te[k] * w1[(long)j * 256 + k];
        hid[j] = siluf(s);
    }
    __syncthreads();
    int cls = y[b];
    for (int j = tid; j < HD; j += 128) {
        float s = b2[j];
        for (int k = 0; k < HD; ++k) s += hid[k] * w2[(long)j * HD + k];
        c[b * HD + j] = s + ytab[(long)cls * HD + j];
    }
}

// ---------------------------------------------------------------------------
// Patch embed: x (B,3,32,32) -> h (B,L,384)
// ---------------------------------------------------------------------------
__global__ void __launch_bounds__(128)
patch_embed_k(const float* __restrict__ x, const float* __restrict__ w1,
              const float* __restrict__ w2, const float* __restrict__ b2,
              const float* __restrict__ pe, float* __restrict__ h)
{
    int tok = blockIdx.x;
    int b = tok >> 8, l = tok & 255;
    int gy = l >> 4, gx = l & 15;
    __shared__ float xp[12];
    __shared__ float inter[128];
    if (threadIdx.x < 12) {
        int cc = threadIdx.x >> 2;
        int py = (threadIdx.x >> 1) & 1;
        int px = threadIdx.x & 1;
        xp[threadIdx.x] = x[(((long)b * 3 + cc) * 32 + (gy * 2 + py)) * 32 + (gx * 2 + px)];
    }
    __syncthreads();
    {
        int m = threadIdx.x;
        float s = 0.f;
        for (int cc = 0; cc < 12; ++cc) s += xp[cc] * w1[m * 12 + cc];
        inter[m] = s;
    }
    __syncthreads();
#pragma unroll
    for (int i = 0; i < 3; ++i) {
        int j = threadIdx.x + i * 128;
        float s = b2[j];
        for (int m = 0; m < 128; ++m) s += inter[m] * w2[(long)j * 128 + m];
        h[(long)tok * HD + j] = s + pe[(long)l * HD + j];
    }
}

// ---------------------------------------------------------------------------
// adaLN modulation: out[b,o] = sum_j silu(c[b,j])*W[o,j] + bias[o]
// ---------------------------------------------------------------------------
__global__ void __launch_bounds__(128)
adaln_k(const float* __restrict__ c, const float* __restrict__ W,
        const float* __restrict__ bias, float* __restrict__ out, int N)
{
    int o = blockIdx.x * 128 + threadIdx.x;
    int b = blockIdx.y;
    __shared__ float cs[HD];
    for (int j = threadIdx.x; j < HD; j += 128) cs[j] = siluf(c[b * HD + j]);
    __syncthreads();
    if (o >= N) return;
    float s = bias[o];
    const float* wr = W + (long)o * HD;
    for (int j = 0; j < HD; j += 4) {
        v4f wv = *(const v4f*)(wr + j);
        s += cs[j] * wv[0] + cs[j + 1] * wv[1] + cs[j + 2] * wv[2] + cs[j + 3] * wv[3];
    }
    out[(long)b * N + o] = s;
}

// ---------------------------------------------------------------------------
// RMS-norm + modulate: out = rms(h)*nw * (1+sc) + sh   (per token)
// ---------------------------------------------------------------------------
__global__ void __launch_bounds__(128)
rms_mod_k(const float* __restrict__ H, const float* __restrict__ nw,
          const float* __restrict__ mod, int moff, float* __restrict__ out)
{
    int tok = blockIdx.x;
    int b = tok >> 8;
    const float* hr = H + (long)tok * HD;
    __shared__ float red[128];
    float v0 = hr[threadIdx.x], v1 = hr[threadIdx.x + 128], v2 = hr[threadIdx.x + 256];
    float ss = blockReduceSum128(v0 * v0 + v1 * v1 + v2 * v2, red);
    float inv = rsqrtf(ss / (float)HD + 1e-6f);
    const float* shv = mod + (long)b * 2304 + moff;
    const float* scv = shv + HD;
    float vv[3] = {v0, v1, v2};
#pragma unroll
    for (int i = 0; i < 3; ++i) {
        int d = threadIdx.x + i * 128;
        out[(long)tok * HD + d] = vv[i] * inv * nw[d] * (1.f + scv[d]) + shv[d];
    }
}

// ---------------------------------------------------------------------------
// Depthwise 3x3 conv + bias + SiLU on z (B,D,16,16); scatter 4 scan directions
// xs layout: (B,4,D,L)
// ---------------------------------------------------------------------------
__global__ void __launch_bounds__(256)
dwconv_scatter_k(const float* __restrict__ z, const float* __restrict__ cw,
                 const float* __restrict__ cb, float* __restrict__ xs)
{
    int i = blockIdx.x * 256 + threadIdx.x;      // B*D*L
    int xx = i & 15, yy = (i >> 4) & 15;
    int d = (i >> 8) % HD;
    int b = i / (HD * LL);
    const float* zp = z + ((long)b * HD + d) * LL;
    float acc = cb[d];
#pragma unroll
    for (int ky = 0; ky < 3; ++ky) {
#pragma unroll
        for (int kx = 0; kx < 3; ++kx) {
            int y2 = yy + ky - 1, x2 = xx + kx - 1;
            if (y2 >= 0 && y2 < 16 && x2 >= 0 && x2 < 16)
                acc += zp[y2 * 16 + x2] * cw[d * 9 + ky * 3 + kx];
        }
    }
    float v = siluf(acc);
    int l = yy * 16 + xx;
    int lt = xx * 16 + yy;
    long base = ((long)b * 4 * HD + d) * LL;
    xs[base + l] = v;
    xs[base + (long)HD * LL + (LL - 1 - l)] = v;
    xs[base + 2L * HD * LL + lt] = v;
    xs[base + 3L * HD * LL + (LL - 1 - lt)] = v;
}

// ---------------------------------------------------------------------------
// Selective scan: one thread per (b,k,d) chain, 16 states in registers.
// xdbl layout is (b,k,l,56): B state at [l*56+24..39], C at [l*56+40..55],
// both 16B-aligned -> b128 loads.
// ---------------------------------------------------------------------------
__global__ void __launch_bounds__(128)
ss_scan_k(const float* __restrict__ dtg, const float* __restrict__ xs,
          const float* __restrict__ xdbl, const float* __restrict__ dtb,
          const float* __restrict__ alog, const float* __restrict__ Ds,
          float* __restrict__ ys)
{
    int d = blockIdx.x * 128 + threadIdx.x;      // 0..383
    int bk = blockIdx.y;                          // b*4+k
    int k = bk & 3;
    long cd = (long)bk * HD + d;
    const float* dtp = dtg + cd * LL;
    const float* xp  = xs + cd * LL;
    const float* xb  = xdbl + (long)bk * LL * 56;
    float* yp = ys + cd * LL;
    long kd = (long)k * HD + d;
    float Ac[NST];
#pragma unroll
    for (int n = 0; n < NST; ++n) Ac[n] = -__expf(alog[kd * NST + n]);
    float db = dtb[kd], Dv = Ds[kd];
    float hst[NST];
#pragma unroll
    for (int n = 0; n < NST; ++n) hst[n] = 0.f;
    for (int l = 0; l < LL; ++l) {
        const float* rowp = xb + (long)l * 56;
        float Ba[NST], Ca[NST];
#pragma unroll
        for (int q = 0; q < 4; ++q) {
            v4f bq = *(const v4f*)(rowp + 24 + 4 * q);
            v4f cq = *(const v4f*)(rowp + 40 + 4 * q);
            Ba[4 * q] = bq[0]; Ba[4 * q + 1] = bq[1]; Ba[4 * q + 2] = bq[2]; Ba[4 * q + 3] = bq[3];
            Ca[4 * q] = cq[0]; Ca[4 * q + 1] = cq[1]; Ca[4 * q + 2] = cq[2]; Ca[4 * q + 3] = cq[3];
        }
        float dtv = dtp[l] + db;
        dtv = (dtv > 20.f) ? dtv : log1pf(__expf(dtv));
        float xv = xp[l];
        float du = dtv * xv;
        float yv = 0.f;
#pragma unroll
        for (int n = 0; n < NST; ++n) {
            hst[n] = __expf(dtv * Ac[n]) * hst[n] + du * Ba[n];
            yv += hst[n] * Ca[n];
        }
        yp[l] = yv + Dv * xv;
    }
}

// ---------------------------------------------------------------------------
// Merge 4 directions + LayerNorm(onorm) -> yln (B,L,D) row-major
// ---------------------------------------------------------------------------
__global__ void __launch_bounds__(128)
merge_ln_k(const float* __restrict__ ys, const float* __restrict__ ow,
           const float* __restrict__ ob, float* __restrict__ yln)
{
    int tok = blockIdx.x;
    int b = tok >> 8, l = tok & 255;
    int lt = ((l & 15) << 4) | (l >> 4);
    __shared__ float red[128];
    float v[3];
#pragma unroll
    for (int i = 0; i < 3; ++i) {
        int d = threadIdx.x + i * 128;
        long b4 = ((long)b * 4 * HD + d) * LL;
        v[i] = ys[b4 + l]
             + ys[b4 + (long)HD * LL + (LL - 1 - l)]
             + ys[b4 + 2L * HD * LL + lt]
             + ys[b4 + 3L * HD * LL + (LL - 1 - lt)];
    }
    float mu = blockReduceSum128(v[0] + v[1] + v[2], red) / (float)HD;
    float q = blockReduceSum128((v[0] - mu) * (v[0] - mu) + (v[1] - mu) * (v[1] - mu) +
                                (v[2] - mu) * (v[2] - mu), red);
    float inv = rsqrtf(q / (float)HD + 1e-5f);
#pragma unroll
    for (int i = 0; i < 3; ++i) {
        int d = threadIdx.x + i * 128;
        yln[(long)tok * HD + d] = (v[i] - mu) * inv * ow[d] + ob[d];
    }
}

// ---------------------------------------------------------------------------
// Gated residual: h += mod[b, goff + d] * v
// ---------------------------------------------------------------------------
__global__ void __launch_bounds__(256)
resid_gate_k(float* __restrict__ H, const float* __restrict__ V,
             const float* __restrict__ mod, int goff)
{
    long i = (long)blockIdx.x * 256 + threadIdx.x;   // B*L*HD
    int b = (int)(i / ((long)LL * HD));
    int d = (int)(i % HD);
    H[i] += mod[(long)b * 2304 + goff + d] * V[i];
}

// ---------------------------------------------------------------------------
// GLU: act = silu(u1) * u2
// ---------------------------------------------------------------------------
__global__ void __launch_bounds__(256)
glu_k(const float* __restrict__ u12, float* __restrict__ act)
{
    long i = (long)blockIdx.x * 256 + threadIdx.x;   // (B*L)*MH
    long m = i / MH_;
    int j = (int)(i % MH_);
    float a = u12[m * (2 * MH_) + j];
    float b = u12[m * (2 * MH_) + MH_ + j];
    act[i] = siluf(a) * b;
}

// ---------------------------------------------------------------------------
// Final: LN (no affine, eps 1e-6) + modulate with (shf, scf)
// ---------------------------------------------------------------------------
__global__ void __launch_bounds__(128)
lnna_mod_k(const float* __restrict__ H, const float* __restrict__ modf,
           float* __restrict__ out)
{
    int tok = blockIdx.x;
    int b = tok >> 8;
    const float* hr = H + (long)tok * HD;
    __shared__ float red[128];
    float v[3] = {hr[threadIdx.x], hr[threadIdx.x + 128], hr[threadIdx.x + 256]};
    float mu = blockReduceSum128(v[0] + v[1] + v[2], red) / (float)HD;
    float q = blockReduceSum128((v[0] - mu) * (v[0] - mu) + (v[1] - mu) * (v[1] - mu) +
                                (v[2] - mu) * (v[2] - mu), red);
    float inv = rsqrtf(q / (float)HD + 1e-6f);
    const float* shf = modf + (long)b * 768;
    const float* scf = shf + HD;
#pragma unroll
    for (int i = 0; i < 3; ++i) {
        int d = threadIdx.x + i * 128;
        out[(long)tok * HD + d] = (v[i] - mu) * inv * (1.f + scf[d]) + shf[d];
    }
}

// ---------------------------------------------------------------------------
// Final projection + un-patchify: out (B,3,32,32); channel j = py*6+px*3+c
// ---------------------------------------------------------------------------
__global__ void __launch_bounds__(128)
final_proj_k(const float* __restrict__ hm, const float* __restrict__ w,
             const float* __restrict__ bias, float* __restrict__ out)
{
    int tok = blockIdx.x;
    int b = tok >> 8, l = tok & 255;
    int gy = l >> 4, gx = l & 15;
    __shared__ float row[HD];
    for (int i = threadIdx.x; i < HD; i += 128) row[i] = hm[(long)tok * HD + i];
    __syncthreads();
    if (threadIdx.x < 12) {
        int j = threadIdx.x;
        float s = bias[j];
        for (int k = 0; k < HD; ++k) s += row[k] * w[(long)j * HD + k];
        int py = j / 6, px = (j / 3) % 2, cc = j % 3;
        out[(((long)b * 3 + cc) * 32 + (gy * 2 + py)) * 32 + (gx * 2 + px)] = s;
    }
}

// ---------------------------------------------------------------------------
// Host orchestration
// ---------------------------------------------------------------------------
static void launch_gemm(hipStream_t st, bool ak1, const float* A, const float* W,
                        const float* bias, float* C, int M, int N, int K,
                        long lam, long lak, long lwn, long lcm, long lcn,
                        int nb,
                        int bA, long sAlo, long sAhi,
                        int bW, long sWlo, long sWhi,
                        int bC, long sClo, long sChi)
{
    dim3 g((M + 63) / 64, (N + 63) / 64, nb);
    bool nal = (N % 64) == 0;
    if (ak1) {
        if (nal)
            gemm_wmma_f32<1, 1><<<g, 128, 0, st>>>(A, W, bias, C, M, N, K, lam, lak, lwn, lcm, lcn,
                                                   bA, sAlo, sAhi, bW, sWlo, sWhi, bC, sClo, sChi);
        else
            gemm_wmma_f32<1, 0><<<g, 128, 0, st>>>(A, W, bias, C, M, N, K, lam, lak, lwn, lcm, lcn,
                                                   bA, sAlo, sAhi, bW, sWlo, sWhi, bC, sClo, sChi);
    } else {
        if (nal)
            gemm_wmma_f32<0, 1><<<g, 128, 0, st>>>(A, W, bias, C, M, N, K, lam, lak, lwn, lcm, lcn,
                                                   bA, sAlo, sAhi, bW, sWlo, sWhi, bC, sClo, sChi);
        else
            gemm_wmma_f32<0, 0><<<g, 128, 0, st>>>(A, W, bias, C, M, N, K, lam, lak, lwn, lcm, lcn,
                                                   bA, sAlo, sAhi, bW, sWlo, sWhi, bC, sClo, sChi);
    }
}

extern "C" void kernel_launch(void* const* d_in, const int* in_sizes, int n_in,
                              void* d_out, int out_size, void* d_ws, size_t ws_size,
                              hipStream_t stream)
{
    (void)in_sizes; (void)n_in; (void)out_size; (void)ws_size;

    const float* x        = (const float*)d_in[0];
    const float* t        = (const float*)d_in[1];
    const int*   yc       = (const int*)  d_in[2];
    const float* t_w1     = (const float*)d_in[3];
    const float* t_b1     = (const float*)d_in[4];
    const float* t_w2     = (const float*)d_in[5];
    const float* t_b2     = (const float*)d_in[6];
    const float* y_table  = (const float*)d_in[7];
    const float* pe_w1    = (const float*)d_in[8];
    const float* pe_w2    = (const float*)d_in[9];
    const float* pe_b2    = (const float*)d_in[10];
    const float* pos_emb  = (const float*)d_in[11];
    const float* adaln_w  = (const float*)d_in[12];
    const float* adaln_b  = (const float*)d_in[13];
    const float* n1_w     = (const float*)d_in[14];
    const float* n2_w     = (const float*)d_in[15];
    const float* in_w     = (const float*)d_in[16];
    const float* conv_w   = (const float*)d_in[17];
    const float* conv_b   = (const float*)d_in[18];
    const float* xproj_w  = (const float*)d_in[19];
    const float* dt_w     = (const float*)d_in[20];
    const float* dt_b     = (const float*)d_in[21];
    const float* A_logs   = (const float*)d_in[22];
    const float* Ds       = (const float*)d_in[23];
    const float* onorm_w  = (const float*)d_in[24];
    const float* onorm_b  = (const float*)d_in[25];
    const float* outp_w   = (const float*)d_in[26];
    const float* mlp_w1   = (const float*)d_in[27];
    const float* mlp_b1   = (const float*)d_in[28];
    const float* mlp_w2   = (const float*)d_in[29];
    const float* mlp_b2   = (const float*)d_in[30];
    const float* f_ada_w  = (const float*)d_in[31];
    const float* f_ada_b  = (const float*)d_in[32];
    const float* f_lin_w  = (const float*)d_in[33];
    const float* f_lin_b  = (const float*)d_in[34];

    // workspace layout
    float* p = (float*)d_ws;
    float* c_emb = p;  p += (long)BB * HD;
    float* h     = p;  p += (long)BB * LL * HD;
    float* xin   = p;  p += (long)BB * LL * HD;
    float* mod   = p;  p += (long)BB * 2304;
    float* z     = p;  p += (long)BB * HD * LL;
    float* xs    = p;  p += (long)BB * 4 * HD * LL;
    float* xdbl  = p;  p += (long)BB * 4 * LL * 56;
    float* dtg   = p;  p += (long)BB * 4 * HD * LL;
    float* ysb   = p;  p += (long)BB * 4 * HD * LL;
    float* yln   = p;  p += (long)BB * LL * HD;
    float* ssout = p;  p += (long)BB * LL * HD;
    float* u12   = p;  p += (long)BB * LL * 2 * MH_;
    float* act   = p;  p += (long)BB * LL * MH_;
    float* mlpo  = p;  p += (long)BB * LL * HD;
    float* modf  = p;  p += (long)BB * 768;
    float* hfin  = p;  p += (long)BB * LL * HD;

    const long NTOK = (long)BB * LL;                  // 1024
    const long ELEMS = NTOK * HD;                     // 393216

    time_embed_k<<<BB, 128, 0, stream>>>(t, yc, t_w1, t_b1, t_w2, t_b2, y_table, c_emb);
    patch_embed_k<<<(int)NTOK, 128, 0, stream>>>(x, pe_w1, pe_w2, pe_b2, pos_emb, h);

    for (int i = 0; i < 8; ++i) {
        const float* aw  = adaln_w + (long)i * 2304 * HD;
        const float* ab  = adaln_b + (long)i * 2304;
        const float* iw  = in_w    + (long)i * HD * HD;
        const float* cwp = conv_w  + (long)i * HD * 9;
        const float* cbp = conv_b  + (long)i * HD;
        const float* xpw = xproj_w + (long)i * 4 * 56 * HD;
        const float* dtw = dt_w    + (long)i * 4 * HD * RANK_;
        const float* dtbp= dt_b    + (long)i * 4 * HD;
        const float* alg = A_logs  + (long)i * 4 * HD * NST;
        const float* dsp = Ds      + (long)i * 4 * HD;
        const float* onw = onorm_w + (long)i * HD;
        const float* onb = onorm_b + (long)i * HD;
        const float* opw = outp_w  + (long)i * HD * HD;
        const float* m1w = mlp_w1  + (long)i * 2 * MH_ * HD;
        const float* m1b = mlp_b1  + (long)i * 2 * MH_;
        const float* m2w = mlp_w2  + (long)i * HD * MH_;
        const float* m2b = mlp_b2  + (long)i * HD;

        adaln_k<<<dim3(18, BB), 128, 0, stream>>>(c_emb, aw, ab, mod, 2304);
        rms_mod_k<<<(int)NTOK, 128, 0, stream>>>(h, n1_w + (long)i * HD, mod, 0, xin);

        // z(b,d,l) = xin(b,l,:) @ in_w^T   (channel-major output for the conv)
        launch_gemm(stream, true, xin, iw, nullptr, z, LL, HD, HD,
                    HD, 1, HD, 1, LL, BB,
                    BB, (long)LL * HD, 0, 1, 0, 0, BB, (long)HD * LL, 0);

        dwconv_scatter_k<<<(int)(ELEMS / 256), 256, 0, stream>>>(z, cwp, cbp, xs);

        // x_dbl(b,k,l,o) = xs(b,k,:,l) @ xproj_w[k]^T  — 16 batches, (l,o) output
        launch_gemm(stream, false, xs, xpw, nullptr, xdbl, LL, 56, HD,
                    1, LL, HD, 56, 1, BB * 4,
                    16, (long)HD * LL, 0, 4, 56L * HD, 0, 16, (long)LL * 56, 0);

        // dt(b,k,d,l) = dtr(b,k,l,:) @ dt_w[k]^T  — dtr = x_dbl cols [0,24)
        launch_gemm(stream, true, xdbl, dtw, nullptr, dtg, LL, HD, RANK_,
                    56, 1, RANK_, 1, LL, BB * 4,
                    16, (long)LL * 56, 0, 4, (long)HD * RANK_, 0, 16, (long)HD * LL, 0);

        ss_scan_k<<<dim3(3, BB * 4), 128, 0, stream>>>(dtg, xs, xdbl, dtbp, alg, dsp, ysb);
        merge_ln_k<<<(int)NTOK, 128, 0, stream>>>(ysb, onw, onb, yln);

        // ssout = yln @ outp_w^T  (row-major, M=1024)
        launch_gemm(stream, true, yln, opw, nullptr, ssout, (int)NTOK, HD, HD,
                    HD, 1, HD, HD, 1, 1,
                    1, 0, 0, 1, 0, 0, 1, 0, 0);
        resid_gate_k<<<(int)(ELEMS / 256), 256, 0, stream>>>(h, ssout, mod, 768);

        // MLP
        rms_mod_k<<<(int)NTOK, 128, 0, stream>>>(h, n2_w + (long)i * HD, mod, 1152, xin);
        launch_gemm(stream, true, xin, m1w, m1b, u12, (int)NTOK, 2 * MH_, HD,
                    HD, 1, HD, 2 * MH_, 1, 1,
                    1, 0, 0, 1, 0, 0, 1, 0, 0);
        glu_k<<<(int)(NTOK * MH_ / 256), 256, 0, stream>>>(u12, act);
        launch_gemm(stream, true, act, m2w, m2b, mlpo, (int)NTOK, HD, MH_,
                    MH_, 1, MH_, HD, 1, 1,
                    1, 0, 0, 1, 0, 0, 1, 0, 0);
        resid_gate_k<<<(int)(ELEMS / 256), 256, 0, stream>>>(h, mlpo, mod, 1920);
    }

    adaln_k<<<dim3(6, BB), 128, 0, stream>>>(c_emb, f_ada_w, f_ada_b, modf, 768);
    lnna_mod_k<<<(int)NTOK, 128, 0, stream>>>(h, modf, hfin);
    final_proj_k<<<(int)NTOK, 128, 0, stream>>>(hfin, f_lin_w, f_lin_b, (float*)d_out);
}